// JumpLinkConv_49658411876805
// MI455X (gfx1250) — compile-verified
//
#include <hip/hip_runtime.h>

// ---------------- problem constants (match reference) ----------------
#define NV    100000     // vertices (== 16 * 6250, exact 16-row tiles)
#define NE    20000      // hyperedges
#define NNZK  1000000    // incidence entries
#define DIM   64
// ALPHA = BETA = 0.5

typedef __attribute__((ext_vector_type(2))) float v2f;
typedef __attribute__((ext_vector_type(8))) float v8f;

// ---------------------------------------------------------------------
// Kernel 0: zero the accumulation buffers (Xe | Xv) living in d_ws.
// ---------------------------------------------------------------------
__global__ __launch_bounds__(256) void zero_ws_kernel(float* __restrict__ ws, long n) {
  long stride = (long)gridDim.x * blockDim.x;
  for (long i = (long)blockIdx.x * blockDim.x + threadIdx.x; i < n; i += stride)
    ws[i] = 0.0f;
}

// ---------------------------------------------------------------------
// Kernel 1: vertex -> edge scatter-add.  Xe[e,:] += X[v,:]
// One thread per (nnz entry, 4-float chunk): 128-bit gather +
// 4 hardware GLOBAL_ATOMIC_ADD_F32 (relaxed, device scope -> L2 atomics).
// ---------------------------------------------------------------------
__global__ __launch_bounds__(256) void scatter_v2e_kernel(
    const float* __restrict__ X, const int* __restrict__ vertex,
    const int* __restrict__ edges, float* __restrict__ Xe) {
  int t = blockIdx.x * 256 + threadIdx.x;
  int i = t >> 4;               // nnz entry
  if (i >= NNZK) return;
  int p = (t & 15) << 2;        // column offset (0..60, step 4)
  int v = vertex[i];
  int e = edges[i];
  const float4 x = *reinterpret_cast<const float4*>(X + (long)v * DIM + p);
  float* dst = Xe + (long)e * DIM + p;
  __hip_atomic_fetch_add(dst + 0, x.x, __ATOMIC_RELAXED, __HIP_MEMORY_SCOPE_AGENT);
  __hip_atomic_fetch_add(dst + 1, x.y, __ATOMIC_RELAXED, __HIP_MEMORY_SCOPE_AGENT);
  __hip_atomic_fetch_add(dst + 2, x.z, __ATOMIC_RELAXED, __HIP_MEMORY_SCOPE_AGENT);
  __hip_atomic_fetch_add(dst + 3, x.w, __ATOMIC_RELAXED, __HIP_MEMORY_SCOPE_AGENT);
}

// ---------------------------------------------------------------------
// Kernel 2: edge -> vertex scatter-add.  Xv[v,:] += Xe[e,:]
// Xe is 5.1 MB -> fully L2 resident, gathers are cheap.
// ---------------------------------------------------------------------
__global__ __launch_bounds__(256) void scatter_e2v_kernel(
    const float* __restrict__ Xe, const int* __restrict__ vertex,
    const int* __restrict__ edges, float* __restrict__ Xv) {
  int t = blockIdx.x * 256 + threadIdx.x;
  int i = t >> 4;
  if (i >= NNZK) return;
  int p = (t & 15) << 2;
  int v = vertex[i];
  int e = edges[i];
  const float4 x = *reinterpret_cast<const float4*>(Xe + (long)e * DIM + p);
  float* dst = Xv + (long)v * DIM + p;
  __hip_atomic_fetch_add(dst + 0, x.x, __ATOMIC_RELAXED, __HIP_MEMORY_SCOPE_AGENT);
  __hip_atomic_fetch_add(dst + 1, x.y, __ATOMIC_RELAXED, __HIP_MEMORY_SCOPE_AGENT);
  __hip_atomic_fetch_add(dst + 2, x.z, __ATOMIC_RELAXED, __HIP_MEMORY_SCOPE_AGENT);
  __hip_atomic_fetch_add(dst + 3, x.w, __ATOMIC_RELAXED, __HIP_MEMORY_SCOPE_AGENT);
}

// ---------------------------------------------------------------------
// Kernel 3: Xi = 0.5*Xv + 0.5*X0 ; h = relu(Xi@W1 + b1) ;
//           out = 0.5*Xi + 0.5*(h@W2 + b2)
// One wave32 per 16-row tile (NV = 16*6250 exactly -> EXEC always all-1s,
// required for WMMA).  Full-fp32 matrix core path: V_WMMA_F32_16X16X4_F32,
// 16 K-steps x 4 N-tiles per GEMM.  A/H tiles staged in LDS; W1/W2 read
// from global (16 KB each, L2-resident, broadcast across all waves).
//
// Fragment layouts (ISA 7.12.2, wave32):
//   C/D  : VGPR r, lane L  -> M = r + 8*(L>=16), N = L & 15
//   A16x4: lane L, v[0..1] -> M = L & 15, K = 2*(L>=16) + {0,1}
//   B4x16: mirrored: lane L, v[0..1] -> N = L & 15, K = 2*(L>=16) + {0,1}
// ---------------------------------------------------------------------
__global__ __launch_bounds__(32) void jumplink_mlp_kernel(
    const float* __restrict__ Xv, const float* __restrict__ X0,
    const float* __restrict__ W1, const float* __restrict__ b1,
    const float* __restrict__ W2, const float* __restrict__ b2,
    float* __restrict__ out) {
  __shared__ float tileA[16][DIM + 4];  // Xi tile (padded vs bank conflicts)
  __shared__ float tileH[16][DIM + 4];  // relu hidden tile

  const int lane = threadIdx.x;         // 0..31 (wave32)
  const int half = lane >> 4;           // 0 | 1
  const int m    = lane & 15;
  const int row0 = blockIdx.x * 16;

  // ---- stage Xi = 0.5*Xv + 0.5*X0 into LDS -------------------------
#pragma unroll 4
  for (int r = 0; r < 16; ++r) {
    const long g = (long)(row0 + r) * DIM;
    tileA[r][lane]      = 0.5f * (Xv[g + lane]      + X0[g + lane]);
    tileA[r][lane + 32] = 0.5f * (Xv[g + lane + 32] + X0[g + lane + 32]);
  }
  __syncthreads();   // single-wave workgroup: hardware NOP, compiler fence

  // ---- GEMM1: h = relu(Xi @ W1 + b1) -------------------------------
  for (int nt = 0; nt < 4; ++nt) {
    const int n = nt * 16 + m;          // output column (C/D layout)
    v8f acc = {};
#pragma unroll
    for (int kk = 0; kk < 16; ++kk) {
      const int klo = kk * 4 + half * 2;
      v2f a, b;
      a.x = tileA[m][klo];
      a.y = tileA[m][klo + 1];
      b.x = W1[klo * DIM + n];
      b.y = W1[(klo + 1) * DIM + n];
      acc = __builtin_amdgcn_wmma_f32_16x16x4_f32(
          false, a, false, b, (short)0, acc, false, false);
    }
    const float bias = b1[n];
#pragma unroll
    for (int r = 0; r < 8; ++r) {
      const int mr = r + 8 * half;
      const float h = acc[r] + bias;
      tileH[mr][n] = h > 0.0f ? h : 0.0f;
    }
  }
  __syncthreads();

  // ---- GEMM2 + jump link: out = 0.5*Xi + 0.5*(h @ W2 + b2) ---------
  for (int nt = 0; nt < 4; ++nt) {
    const int n = nt * 16 + m;
    v8f acc = {};
#pragma unroll
    for (int kk = 0; kk < 16; ++kk) {
      const int klo = kk * 4 + half * 2;
      v2f a, b;
      a.x = tileH[m][klo];
      a.y = tileH[m][klo + 1];
      b.x = W2[klo * DIM + n];
      b.y = W2[(klo + 1) * DIM + n];
      acc = __builtin_amdgcn_wmma_f32_16x16x4_f32(
          false, a, false, b, (short)0, acc, false, false);
    }
    const float bias = b2[n];
#pragma unroll
    for (int r = 0; r < 8; ++r) {
      const int mr = r + 8 * half;
      const float mlp = acc[r] + bias;
      out[(long)(row0 + mr) * DIM + n] = 0.5f * tileA[mr][n] + 0.5f * mlp;
    }
  }
}

// ---------------------------------------------------------------------
// launch wrapper
// inputs: X, X0, W1, b1, W2, b2, vertex, edges   (reference dict order)
// d_ws layout: [ Xe : NE*DIM f32 | Xv : NV*DIM f32 ]  (= 30.72 MB)
// ---------------------------------------------------------------------
extern "C" void kernel_launch(void* const* d_in, const int* in_sizes, int n_in,
                              void* d_out, int out_size, void* d_ws, size_t ws_size,
                              hipStream_t stream) {
  const float* X      = (const float*)d_in[0];
  const float* X0     = (const float*)d_in[1];
  const float* W1     = (const float*)d_in[2];
  const float* b1     = (const float*)d_in[3];
  const float* W2     = (const float*)d_in[4];
  const float* b2     = (const float*)d_in[5];
  const int*   vertex = (const int*)d_in[6];
  const int*   edges  = (const int*)d_in[7];
  float*       outp   = (float*)d_out;

  float* Xe = (float*)d_ws;                 // NE*DIM
  float* Xv = Xe + (long)NE * DIM;          // NV*DIM

  const long accum_elems = (long)(NE + NV) * DIM;
  zero_ws_kernel<<<2048, 256, 0, stream>>>((float*)d_ws, accum_elems);

  const int scat_blocks = (NNZK * 16 + 255) / 256;  // 62500
  scatter_v2e_kernel<<<scat_blocks, 256, 0, stream>>>(X, vertex, edges, Xe);
  scatter_e2v_kernel<<<scat_blocks, 256, 0, stream>>>(Xe, vertex, edges, Xv);

  jumplink_mlp_kernel<<<NV / 16, 32, 0, stream>>>(Xv, X0, W1, b1, W2, b2, outp);
}